// PEAGCNChannel_55078660604180
// MI455X (gfx1250) — compile-verified
//
#include <hip/hip_runtime.h>
#include <math.h>

typedef float v2f __attribute__((ext_vector_type(2)));
typedef float v8f __attribute__((ext_vector_type(8)));

#define EMB 128
#define HID 128
#define REPR 64

// Native fp32 global atomic add (global_atomic_add_f32), avoiding any CAS
// fallback in the dominant scatter phase.
__device__ __forceinline__ void atomic_add_f32(float* p, float v) {
#if defined(__HIP_DEVICE_COMPILE__)
  unsafeAtomicAdd(p, v);
#else
  atomicAdd(p, v);
#endif
}

// ---------------------------------------------------------------------------
// WMMA f32 GEMM:  C0[M x NT*16] = A[M x K] @ B[K x NT*16] (+bias0)
// optional fused second output: C1 = (A@B) * dinv[row]^2 + bias1[col]
// One wave computes a full 16-row x NT*16-col slab. blockDim = 256 (8 waves).
// Uses V_WMMA_F32_16X16X4_F32 (CDNA5, wave32).
//
// Fragment layouts per CDNA5 ISA 7.12.2:
//  A 16x4 f32: lane l (l<16): M=l, v0=K, v1=K+1 ; lane l+16: K+2, K+3
//  B 4x16 f32: lane l (l<16): N=l, v0=row K, v1=row K+1 ; lane l+16: K+2, K+3
//  C/D 16x16 f32: VGPR r: lanes 0-15 -> M=r, N=lane ; lanes 16-31 -> M=r+8
//
// A-row loads are clamped (not predicated) so the K-loop stays branch-free:
// no exec-mask save/restore, single loop body, EXEC all-ones for WMMA.
// ---------------------------------------------------------------------------
template <int NT>
__global__ __launch_bounds__(256) void gemm_wmma(
    const float* __restrict__ A, int lda, const float* __restrict__ B, int K,
    int M, const float* __restrict__ bias0, float* __restrict__ C0, int ldc0,
    const float* __restrict__ dinv, const float* __restrict__ bias1,
    float* __restrict__ C1) {
  const int ldb = NT * 16;
  const int wave = threadIdx.x >> 5;
  const int lane = threadIdx.x & 31;
  const int lh = lane & 15;   // low half index (M for A, N for B/D)
  const int hh = lane >> 4;   // lane-half selector (K offset 0 or 2)
  const int row0 = (blockIdx.x * 8 + wave) * 16;
  if (row0 >= M) return;  // whole-wave exit: EXEC stays all-ones for WMMA

  int arow = row0 + lh;
  arow = arow < M ? arow : (M - 1);  // branch-free clamp (padding never stored)
  const float* Ap = A + (size_t)arow * lda;

  v8f acc[NT];
#pragma unroll
  for (int j = 0; j < NT; ++j) acc[j] = (v8f)(0.0f);

  for (int k = 0; k < K; k += 4) {
    const int ka = k + 2 * hh;
    const v2f a = *(const v2f*)(Ap + ka);  // 8B load: A[m][ka], A[m][ka+1]
    const float* B0 = B + (size_t)ka * ldb + lh;        // row ka
    const float* B1 = B0 + ldb;                         // row ka+1
#pragma unroll
    for (int j = 0; j < NT; ++j) {
      v2f b;
      b.x = B0[j * 16];
      b.y = B1[j * 16];
      acc[j] = __builtin_amdgcn_wmma_f32_16x16x4_f32(
          false, a, false, b, (short)0, acc[j], false, false);
    }
  }

#pragma unroll
  for (int j = 0; j < NT; ++j) {
    const int colg = j * 16 + lh;
    const float b0 = bias0 ? bias0[colg] : 0.0f;
    const float b1 = bias1 ? bias1[colg] : 0.0f;
#pragma unroll
    for (int r = 0; r < 8; ++r) {
      const int row = row0 + r + 8 * hh;
      if (row < M) {
        const float v = acc[j][r];
        C0[(size_t)row * ldc0 + colg] = v + b0;
        if (C1) {
          const float dv = dinv[row];
          C1[(size_t)row * ldb + colg] = v * dv * dv + b1;
        }
      }
    }
  }
}

// ---------------------------------------------------------------------------
// Degree / normalization
// ---------------------------------------------------------------------------
__global__ __launch_bounds__(256) void init_deg(float* __restrict__ deg, int n) {
  int i = blockIdx.x * 256 + threadIdx.x;
  if (i < n) deg[i] = 1.0f;  // self-loop contributes 1
}

__global__ __launch_bounds__(256) void count_deg(
    const long long* __restrict__ eidx, float* __restrict__ deg, int E) {
  int e = blockIdx.x * 256 + threadIdx.x;
  if (e < E) atomic_add_f32(&deg[(int)eidx[(size_t)E + e]], 1.0f);  // col index
}

__global__ __launch_bounds__(256) void deg_to_dinv(float* __restrict__ d, int n) {
  int i = blockIdx.x * 256 + threadIdx.x;
  if (i < n) d[i] = rsqrtf(d[i]);
}

// ---------------------------------------------------------------------------
// Edge scatter: one wave32 per edge; lane l moves features [4l..4l+3].
// agg[col] += hW[row] * dinv[row]*dinv[col]
// hW (51MB) + agg (51MB) both fit in the 192MB L2 -> runs at L2/atomic rates.
// ---------------------------------------------------------------------------
__global__ __launch_bounds__(256) void scatter_edges(
    const float* __restrict__ hW, const float* __restrict__ dinv,
    const long long* __restrict__ eidx, float* __restrict__ agg, int E) {
  const int wid = blockIdx.x * 8 + (threadIdx.x >> 5);
  if (wid >= E) return;
  const int lane = threadIdx.x & 31;
  const int r = (int)eidx[wid];
  const int c = (int)eidx[(size_t)E + wid];
  const float w = dinv[r] * dinv[c];
  const float4 v = *((const float4*)(hW + (size_t)r * HID) + lane);
  float* dst = agg + (size_t)c * HID + lane * 4;
  atomic_add_f32(dst + 0, v.x * w);
  atomic_add_f32(dst + 1, v.y * w);
  atomic_add_f32(dst + 2, v.z * w);
  atomic_add_f32(dst + 3, v.w * w);
}

// ---------------------------------------------------------------------------
// ReLU into the 384-wide JK-concat buffer (free concatenation via stride)
// ---------------------------------------------------------------------------
__global__ __launch_bounds__(256) void relu_copy(
    const float* __restrict__ agg, float* __restrict__ dst, int M) {
  int i = blockIdx.x * 256 + threadIdx.x;
  if (i >= M * HID) return;
  const int row = i >> 7;
  const int f = i & 127;
  dst[(size_t)row * (3 * HID) + f] = fmaxf(agg[i], 0.0f);
}

// ---------------------------------------------------------------------------
extern "C" void kernel_launch(void* const* d_in, const int* in_sizes, int n_in,
                              void* d_out, int out_size, void* d_ws,
                              size_t ws_size, hipStream_t stream) {
  const float* x = (const float*)d_in[0];
  const long long* eidx = (const long long*)d_in[1];
  const float* W[3] = {(const float*)d_in[2], (const float*)d_in[4],
                       (const float*)d_in[6]};
  const float* b[3] = {(const float*)d_in[3], (const float*)d_in[5],
                       (const float*)d_in[7]};
  const float* Wlin = (const float*)d_in[8];
  const float* blin = (const float*)d_in[9];

  const int N = in_sizes[0] / EMB;   // 100000
  const int E = in_sizes[1] / 2;     // 1600000

  // workspace carve-up (floats): dinv | hW | agg | hcat
  float* dinv = (float*)d_ws;
  float* hW = dinv + (((size_t)N + 255) & ~(size_t)255);
  float* agg = hW + (size_t)N * HID;
  float* hcat = agg + (size_t)N * HID;  // [N, 3*HID]

  // degree -> dinv
  init_deg<<<(N + 255) / 256, 256, 0, stream>>>(dinv, N);
  count_deg<<<(E + 255) / 256, 256, 0, stream>>>(eidx, dinv, E);
  deg_to_dinv<<<(N + 255) / 256, 256, 0, stream>>>(dinv, N);

  const int gemm_blocks = (N + 127) / 128;  // 128 rows per block (8 waves)

  const float* Hin = x;
  int lda = EMB;
  for (int l = 0; l < 3; ++l) {
    // hW = Hin @ W[l];  agg = hW * dinv^2 + b[l]   (fused epilogue)
    gemm_wmma<8><<<gemm_blocks, 256, 0, stream>>>(
        Hin, lda, W[l], HID, N, nullptr, hW, HID, dinv, b[l], agg);
    // agg[col] += hW[row] * norm(e)
    scatter_edges<<<(E + 7) / 8, 256, 0, stream>>>(hW, dinv, eidx, agg, E);
    // hcat[:, l*128:(l+1)*128] = relu(agg)
    relu_copy<<<((size_t)N * HID + 255) / 256, 256, 0, stream>>>(
        agg, hcat + l * HID, N);
    Hin = hcat + l * HID;
    lda = 3 * HID;
  }

  // out = hcat @ Wlin + blin   (K = 384, 4 column tiles of 16)
  gemm_wmma<4><<<gemm_blocks, 256, 0, stream>>>(
      hcat, 3 * HID, Wlin, 3 * HID, N, blin, (float*)d_out, REPR, nullptr,
      nullptr, nullptr);
}